// GCN_70274254897749
// MI455X (gfx1250) — compile-verified
//
#include <hip/hip_runtime.h>
#include <hip/hip_bf16.h>

// ---------------------------------------------------------------------------
// GCN forward (2x GCNConv + FC) for MI455X / gfx1250, wave32 + WMMA f32.
//
// Problem sizes (fixed by the reference):
constexpr int N_NODES = 100000;
constexpr int N_EDGES = 600000;
constexpr int N_CLASS = 26;
constexpr int HIDDEN  = 128;
//
// Bound analysis: edge gather/scatter moves ~1.4 GB (dominant, ~60us at
// 23.3 TB/s, mostly L2-resident since working set ~100 MB << 192 MB L2);
// GEMMs are ~5 GFLOP total -> V_WMMA_F32_16X16X4_F32 (exact fp32 on the
// matrix pipe). Each wave owns two 16x16 N-tiles and reuses its A fragment:
// inner loop = 3x ds_load_b64 : 2x v_wmma.
// ---------------------------------------------------------------------------

typedef float v2f __attribute__((ext_vector_type(2)));
typedef float v8f __attribute__((ext_vector_type(8)));

// ------------------------------ norm kernels -------------------------------

__global__ void fill_ones_kernel(float* __restrict__ p, int n) {
    int i = blockIdx.x * blockDim.x + threadIdx.x;
    if (i < n) p[i] = 1.0f;   // self-loop weight contributes 1 to each degree
}

__global__ void deg_accum_kernel(float* __restrict__ deg,
                                 const int* __restrict__ col,
                                 const float* __restrict__ w, int E) {
    int e = blockIdx.x * blockDim.x + threadIdx.x;
    if (e < E) atomicAdd(&deg[col[e]], w[e]);
}

__global__ void deg_to_dinv_kernel(float* __restrict__ d, int n) {
    int i = blockIdx.x * blockDim.x + threadIdx.x;
    if (i < n) {
        float v = d[i];
        d[i] = (v > 0.0f) ? rsqrtf(fmaxf(v, 1e-12f)) : 0.0f;
    }
}

__global__ void edge_norm_kernel(float* __restrict__ norm,
                                 const int* __restrict__ row,
                                 const int* __restrict__ col,
                                 const float* __restrict__ w,
                                 const float* __restrict__ dinv, int E) {
    int e = blockIdx.x * blockDim.x + threadIdx.x;
    if (e < E) norm[e] = dinv[row[e]] * w[e] * dinv[col[e]];
}

// ------------------------------ WMMA GEMM ----------------------------------
// C[M x N] = op(A)[M x K] @ W[K x N] (+ bias), op = optional ReLU on A reads.
// Block: 16 output rows; each wave computes TWO 16x16 N-tiles (A reuse).
// K staged through LDS in chunks of KC. W is staged K-pair-interleaved so a
// lane's B fragment (two consecutive K at one column) is one ds_load_b64:
//   sW[ ((k>>1)*Npad + n)*2 + (k&1) ] = W[k][n]

constexpr int KC   = 64;    // K-chunk staged in LDS (multiple of 4)
constexpr int NMAX = 128;   // max padded N handled per block

template <bool RELU_A, bool HAS_BIAS>
__global__ void gcn_wmma_gemm_kernel(const float* __restrict__ A,
                                     const float* __restrict__ W,
                                     const float* __restrict__ bias,
                                     float* __restrict__ C,
                                     int M, int K, int N) {
    __shared__ float sW[KC * NMAX];   // paired layout, 32 KB max
    __shared__ float sA[16 * KC];     // [16][kc],      4 KB max

    const int Npad    = ((int)blockDim.x >> 5) << 5;  // 32 cols per wave
    const int Kpad    = (K + 3) & ~3;
    const int rowBase = blockIdx.x << 4;
    const int tid     = threadIdx.x;
    const int nth     = blockDim.x;
    const int lane    = tid & 31;
    const int wave    = tid >> 5;
    const int lane15  = lane & 15;
    const int khalf   = (lane >> 4) << 1;  // 0 or 2: K-pair owned by lane half
    const int nTile0  = wave << 5;         // first of this wave's two N-tiles

    v8f acc0 = {};
    v8f acc1 = {};

    for (int kb = 0; kb < Kpad; kb += KC) {
        const int kc = (Kpad - kb < KC) ? (Kpad - kb) : KC;  // multiple of 4
        __syncthreads();
        // Stage W[kb .. kb+kc) x Npad into paired layout, zero-padded.
        for (int i = tid; i < kc * Npad; i += nth) {
            int kk = i / Npad;
            int n  = i - kk * Npad;
            int k  = kb + kk;
            float v = (k < K && n < N) ? W[(size_t)k * N + n] : 0.0f;
            sW[(((kk >> 1) * Npad) + n) * 2 + (kk & 1)] = v;
        }
        // Stage A strip: 16 rows x kc, zero-padded, optional fused ReLU.
        for (int i = tid; i < 16 * kc; i += nth) {
            int r = i / kc;
            int k = kb + (i - r * kc);
            int m = rowBase + r;
            float a = (m < M && k < K) ? A[(size_t)m * K + k] : 0.0f;
            if (RELU_A) a = fmaxf(a, 0.0f);
            sA[i] = a;
        }
        // Prefetch next K-chunk of W into cache while this chunk computes.
        if (kb + KC < K) {
            __builtin_prefetch(W + (size_t)(kb + KC) * N + tid, 0, 0);
        }
        __syncthreads();

#pragma unroll 4
        for (int k = 0; k < kc; k += 4) {
            // A 16x4 fragment: lane half owns K-pair {k+khalf, k+khalf+1}.
            // 8-byte aligned (kc multiple of 4, k+khalf even) -> ds_load_b64.
            v2f a = *(const v2f*)&sA[lane15 * kc + k + khalf];
            // B 4x16 fragments for the wave's two tiles; paired layout makes
            // each a contiguous 8-byte ds_load_b64.
            const float* pb =
                &sW[(((k + khalf) >> 1) * Npad + nTile0 + lane15) * 2];
            v2f b0 = *(const v2f*)pb;
            v2f b1 = *(const v2f*)(pb + 32);   // +16 columns * 2
            acc0 = __builtin_amdgcn_wmma_f32_16x16x4_f32(
                false, a, false, b0, (short)0, acc0, false, false);
            acc1 = __builtin_amdgcn_wmma_f32_16x16x4_f32(
                false, a, false, b1, (short)0, acc1, false, false);
        }
    }

    // C/D layout: VGPR v holds M = v (lanes 0-15) / v+8 (lanes 16-31).
    const int  mhalf  = (lane >> 4) << 3;           // 0 or 8
    const bool fullM  = (rowBase + 16 <= M);        // whole-tile fast path
    const int  n0     = nTile0 + lane15;
    const int  n1     = n0 + 16;

    if (n0 < N) {
        const float bv = HAS_BIAS ? bias[n0] : 0.0f;
        if (fullM) {
#pragma unroll
            for (int v = 0; v < 8; ++v)
                C[(size_t)(rowBase + mhalf + v) * N + n0] = acc0[v] + bv;
        } else {
#pragma unroll
            for (int v = 0; v < 8; ++v) {
                int m = rowBase + mhalf + v;
                if (m < M) C[(size_t)m * N + n0] = acc0[v] + bv;
            }
        }
    }
    if (n1 < N) {
        const float bv = HAS_BIAS ? bias[n1] : 0.0f;
        if (fullM) {
#pragma unroll
            for (int v = 0; v < 8; ++v)
                C[(size_t)(rowBase + mhalf + v) * N + n1] = acc1[v] + bv;
        } else {
#pragma unroll
            for (int v = 0; v < 8; ++v) {
                int m = rowBase + mhalf + v;
                if (m < M) C[(size_t)m * N + n1] = acc1[v] + bv;
            }
        }
    }
}

// --------------------------- edge aggregation ------------------------------
// out[n][:] = dinv[n]^2 * h[n][:] + b[:]              (self loop + bias)
// out[col[e]][:] += norm[e] * h[row[e]][:]            (atomic scatter-add)

__global__ void agg_self_bias_kernel(float* __restrict__ out,
                                     const float* __restrict__ h,
                                     const float* __restrict__ dinv,
                                     const float* __restrict__ bias, int Nn) {
    int t = blockIdx.x * blockDim.x + threadIdx.x;
    const int perRow = HIDDEN / 4;                 // 32 float4 per row
    if (t >= Nn * perRow) return;
    int n = t / perRow;
    int f = (t - n * perRow) * 4;
    float s = dinv[n];
    s = s * s;
    const float4 hv = *(const float4*)(h + (size_t)n * HIDDEN + f);
    const float4 bv = *(const float4*)(bias + f);
    float4 o;
    o.x = fmaf(s, hv.x, bv.x);
    o.y = fmaf(s, hv.y, bv.y);
    o.z = fmaf(s, hv.z, bv.z);
    o.w = fmaf(s, hv.w, bv.w);
    *(float4*)(out + (size_t)n * HIDDEN + f) = o;
}

__global__ void agg_edges_kernel(float* __restrict__ out,
                                 const float* __restrict__ h,
                                 const int* __restrict__ row,
                                 const int* __restrict__ col,
                                 const float* __restrict__ norm, int E) {
    // 32 lanes cooperate on one edge -> e is wave-uniform (scalarizable).
    int e = blockIdx.x * (blockDim.x >> 5) + (threadIdx.x >> 5);
    if (e >= E) return;
    int lane = threadIdx.x & 31;
    int f    = lane * 4;
    int r = row[e];
    int c = col[e];
    float w = norm[e];
    const float4 hv = *(const float4*)(h + (size_t)r * HIDDEN + f);
    float* dst = out + (size_t)c * HIDDEN + f;
    atomicAdd(dst + 0, w * hv.x);
    atomicAdd(dst + 1, w * hv.y);
    atomicAdd(dst + 2, w * hv.z);
    atomicAdd(dst + 3, w * hv.w);
}

// ------------------------------- launcher ----------------------------------

static inline size_t align256(size_t x) { return (x + 255) & ~(size_t)255; }

extern "C" void kernel_launch(void* const* d_in, const int* in_sizes, int n_in,
                              void* d_out, int out_size, void* d_ws, size_t ws_size,
                              hipStream_t stream) {
    (void)in_sizes; (void)n_in; (void)out_size; (void)ws_size;

    const float* x   = (const float*)d_in[0];
    const int*   ei  = (const int*)d_in[1];          // [2, E] flat
    const float* ew  = (const float*)d_in[2];
    const float* W1  = (const float*)d_in[3];
    const float* b1  = (const float*)d_in[4];
    const float* W2  = (const float*)d_in[5];
    const float* b2  = (const float*)d_in[6];
    const float* Wfc = (const float*)d_in[7];
    const float* bfc = (const float*)d_in[8];
    float*       out = (float*)d_out;

    const int* rowIdx = ei;             // source
    const int* colIdx = ei + N_EDGES;   // destination

    // Workspace carve-up (~106 MB).
    char* ws = (char*)d_ws;
    size_t off = 0;
    float* dinv = (float*)(ws + off); off += align256(sizeof(float) * N_NODES);
    float* enrm = (float*)(ws + off); off += align256(sizeof(float) * N_EDGES);
    float* bufA = (float*)(ws + off); off += align256(sizeof(float) * (size_t)N_NODES * HIDDEN);
    float* bufB = (float*)(ws + off); off += align256(sizeof(float) * (size_t)N_NODES * HIDDEN);

    const int TB = 256;
    const int gN  = (N_NODES + TB - 1) / TB;
    const int gE  = (N_EDGES + TB - 1) / TB;
    const int gNH = (N_NODES * (HIDDEN / 4) + TB - 1) / TB;
    const int gEH = (N_EDGES * 32 + TB - 1) / TB;
    const int gM  = (N_NODES + 15) / 16;             // 6250 row-strips

    // Block sizes: one wave per 32 output columns (padded).
    const int thrHidden = 32 * (((HIDDEN + 31) / 32));   // Npad=128 -> 128 thr
    const int thrClass  = 32 * (((N_CLASS + 31) / 32));  // Npad=32  -> 32 thr

    // 1) gcn_norm
    fill_ones_kernel<<<gN, TB, 0, stream>>>(dinv, N_NODES);
    deg_accum_kernel<<<gE, TB, 0, stream>>>(dinv, colIdx, ew, N_EDGES);
    deg_to_dinv_kernel<<<gN, TB, 0, stream>>>(dinv, N_NODES);
    edge_norm_kernel<<<gE, TB, 0, stream>>>(enrm, rowIdx, colIdx, ew, dinv, N_EDGES);

    // 2) layer 1: T = x @ W1 ; H1 = agg(T) + b1   (relu deferred into GEMM2)
    gcn_wmma_gemm_kernel<false, false><<<gM, thrHidden, 0, stream>>>(
        x, W1, nullptr, bufA, N_NODES, N_CLASS, HIDDEN);
    agg_self_bias_kernel<<<gNH, TB, 0, stream>>>(bufB, bufA, dinv, b1, N_NODES);
    agg_edges_kernel<<<gEH, TB, 0, stream>>>(bufB, bufA, rowIdx, colIdx, enrm, N_EDGES);

    // 3) layer 2: T = relu(H1) @ W2 ; H2 = agg(T) + b2   (relu deferred)
    gcn_wmma_gemm_kernel<true, false><<<gM, thrHidden, 0, stream>>>(
        bufB, W2, nullptr, bufA, N_NODES, HIDDEN, HIDDEN);
    agg_self_bias_kernel<<<gNH, TB, 0, stream>>>(bufB, bufA, dinv, b2, N_NODES);
    agg_edges_kernel<<<gEH, TB, 0, stream>>>(bufB, bufA, rowIdx, colIdx, enrm, N_EDGES);

    // 4) FC head: out = relu(H2) @ Wfc + bfc   (Npad=32 -> 1 wave/block)
    gcn_wmma_gemm_kernel<true, true><<<gM, thrClass, 0, stream>>>(
        bufB, Wfc, bfc, out, N_NODES, HIDDEN, N_CLASS);
}